// DAS_15058155340289
// MI455X (gfx1250) — compile-verified
//
#include <hip/hip_runtime.h>
#include <hip/hip_bf16.h>
#include <cstdint>

#define B_   2
#define K_   8
#define NC_  64
#define NS_  2048
#define NZ_  256
#define NX_  256
#define CCHUNK 8                  // channels staged per LDS chunk
#define NCHUNK (NC_ / CCHUNK)     // 8 chunks

// ---------------- CDNA5 async global->LDS helpers ----------------
#if __has_builtin(__builtin_amdgcn_global_load_async_to_lds_b128)
#define HAVE_ASYNC_B128 1
#endif

typedef int v4i_ __attribute__((vector_size(4 * sizeof(int))));
typedef __attribute__((address_space(1))) v4i_* gas_v4i;   // global
typedef __attribute__((address_space(3))) v4i_* las_v4i;   // LDS

__device__ __forceinline__ void async_copy_b128(const float* gsrc, float* ldst) {
#if defined(__HIP_DEVICE_COMPILE__)
#if defined(HAVE_ASYNC_B128)
    __builtin_amdgcn_global_load_async_to_lds_b128(
        (gas_v4i)(gsrc),
        (las_v4i)(ldst),
        0, 0);
#else
    unsigned lo = (unsigned)(uintptr_t)ldst;           // LDS byte address (low 32 bits)
    uint64_t ga = (uint64_t)(uintptr_t)gsrc;
    asm volatile("global_load_async_to_lds_b128 %0, %1, off"
                 :: "v"(lo), "v"(ga) : "memory");
#endif
#else
    (void)gsrc; (void)ldst;
#endif
}

#if __has_builtin(__builtin_amdgcn_s_wait_asynccnt)
#define WAIT_ASYNC(n) __builtin_amdgcn_s_wait_asynccnt(n)
#else
#define WAIT_ASYNC(n) asm volatile("s_wait_asynccnt " #n ::: "memory")
#endif

// ---------------- main DAS kernel (transposed rfs layout) ----------------
// grid = B*NZ blocks, block = NX threads (one pixel per thread, 8 wave32/block).
// Double-buffered async prefetch of samples_idx chunks (8 channels x 256 px = 8KB)
// into LDS while the random 64B tap-pair gathers for the current chunk run.
__global__ __launch_bounds__(256) void das_kernel_trans(
        const float* __restrict__ rfsT,  // [B,NC,NS,K]
        const int*   __restrict__ ids,   // [B]
        const float* __restrict__ sidx,  // [N_IDS,NC,NZ,NX]
        float*       __restrict__ out) { // [B,NZ,NX,K]
    __shared__ float buf[2][CCHUNK * NX_];   // 16 KB double buffer

    const int t = threadIdx.x;               // pixel x
    const int z = blockIdx.x & (NZ_ - 1);
    const int b = blockIdx.x >> 8;           // NZ_ == 256
    const int id = ids[b];

    const float* idx_base = sidx + (size_t)id * NC_ * NZ_ * NX_ + (size_t)z * NX_;
    const size_t cstride  = (size_t)NZ_ * NX_;   // channel stride in samples_idx

    // async-issue mapping: 32 lanes per staged row, 8 floats (2 x b128) per lane
    const int r  = t >> 5;          // 0..7  : channel row within chunk
    const int xo = (t & 31) << 3;   // 0..248: x offset handled by this lane

    // prefetch chunk 0
    {
        const float* g = idx_base + (size_t)r * cstride + xo;
        async_copy_b128(g,     &buf[0][r * NX_ + xo]);
        async_copy_b128(g + 4, &buf[0][r * NX_ + xo + 4]);
    }

    float acc[K_];
#pragma unroll
    for (int k = 0; k < K_; ++k) acc[k] = 0.0f;

    for (int ch = 0; ch < NCHUNK; ++ch) {
        const int cur = ch & 1;
        if (ch + 1 < NCHUNK) {
            const float* g = idx_base + ((size_t)(ch + 1) * CCHUNK + r) * cstride + xo;
            async_copy_b128(g,     &buf[1 - cur][r * NX_ + xo]);
            async_copy_b128(g + 4, &buf[1 - cur][r * NX_ + xo + 4]);
            WAIT_ASYNC(2);   // async loads complete in order: <=2 pending => current chunk landed
        } else {
            WAIT_ASYNC(0);
        }
        __syncthreads();     // current chunk visible to all waves

#pragma unroll
        for (int cl = 0; cl < CCHUNK; ++cl) {
            const int c = ch * CCHUNK + cl;
            float fidx = buf[cur][cl * NX_ + t];
            float fi0  = floorf(fidx);
            int   i0   = (int)fi0;
            float w    = fidx - fi0;
            // per-tap zero padding (matches reference valid-mask semantics)
            float wl = (i0 >= 0  && i0 <= NS_ - 1) ? (1.0f - w) : 0.0f;
            float wr = (i0 >= -1 && i0 <= NS_ - 2) ? w : 0.0f;
            int il = min(max(i0, 0), NS_ - 1);
            int ir = min(max(i0 + 1, 0), NS_ - 1);

            const float* bl = rfsT + (((size_t)b * NC_ + c) * NS_ + il) * K_;
            const float* br = rfsT + (((size_t)b * NC_ + c) * NS_ + ir) * K_;
            float4 l0 = reinterpret_cast<const float4*>(bl)[0];
            float4 l1 = reinterpret_cast<const float4*>(bl)[1];
            float4 r0 = reinterpret_cast<const float4*>(br)[0];
            float4 r1 = reinterpret_cast<const float4*>(br)[1];
            acc[0] += wl * l0.x + wr * r0.x;
            acc[1] += wl * l0.y + wr * r0.y;
            acc[2] += wl * l0.z + wr * r0.z;
            acc[3] += wl * l0.w + wr * r0.w;
            acc[4] += wl * l1.x + wr * r1.x;
            acc[5] += wl * l1.y + wr * r1.y;
            acc[6] += wl * l1.z + wr * r1.z;
            acc[7] += wl * l1.w + wr * r1.w;
        }
        __syncthreads();     // everyone done with buf[cur] before it is overwritten
    }

    float* o = out + (((size_t)b * NZ_ + z) * NX_ + t) * K_;
    float4 o0 = { acc[0], acc[1], acc[2], acc[3] };
    float4 o1 = { acc[4], acc[5], acc[6], acc[7] };
    reinterpret_cast<float4*>(o)[0] = o0;
    reinterpret_cast<float4*>(o)[1] = o1;
}

// ---------------- fallback DAS kernel (original rfs layout) ----------------
__global__ __launch_bounds__(256) void das_kernel_plain(
        const float* __restrict__ rfs,   // [B,K,NC,NS]
        const int*   __restrict__ ids,
        const float* __restrict__ sidx,
        float*       __restrict__ out) {
    __shared__ float buf[2][CCHUNK * NX_];

    const int t = threadIdx.x;
    const int z = blockIdx.x & (NZ_ - 1);
    const int b = blockIdx.x >> 8;
    const int id = ids[b];

    const float* idx_base = sidx + (size_t)id * NC_ * NZ_ * NX_ + (size_t)z * NX_;
    const size_t cstride  = (size_t)NZ_ * NX_;
    const int r  = t >> 5;
    const int xo = (t & 31) << 3;

    {
        const float* g = idx_base + (size_t)r * cstride + xo;
        async_copy_b128(g,     &buf[0][r * NX_ + xo]);
        async_copy_b128(g + 4, &buf[0][r * NX_ + xo + 4]);
    }

    float acc[K_];
#pragma unroll
    for (int k = 0; k < K_; ++k) acc[k] = 0.0f;

    for (int ch = 0; ch < NCHUNK; ++ch) {
        const int cur = ch & 1;
        if (ch + 1 < NCHUNK) {
            const float* g = idx_base + ((size_t)(ch + 1) * CCHUNK + r) * cstride + xo;
            async_copy_b128(g,     &buf[1 - cur][r * NX_ + xo]);
            async_copy_b128(g + 4, &buf[1 - cur][r * NX_ + xo + 4]);
            WAIT_ASYNC(2);
        } else {
            WAIT_ASYNC(0);
        }
        __syncthreads();

#pragma unroll
        for (int cl = 0; cl < CCHUNK; ++cl) {
            const int c = ch * CCHUNK + cl;
            float fidx = buf[cur][cl * NX_ + t];
            float fi0  = floorf(fidx);
            int   i0   = (int)fi0;
            float w    = fidx - fi0;
            float wl = (i0 >= 0  && i0 <= NS_ - 1) ? (1.0f - w) : 0.0f;
            float wr = (i0 >= -1 && i0 <= NS_ - 2) ? w : 0.0f;
            int il = min(max(i0, 0), NS_ - 1);
            int ir = min(max(i0 + 1, 0), NS_ - 1);
#pragma unroll
            for (int k = 0; k < K_; ++k) {
                const float* base = rfs + (((size_t)b * K_ + k) * NC_ + c) * NS_;
                acc[k] += wl * base[il] + wr * base[ir];
            }
        }
        __syncthreads();
    }

    float* o = out + (((size_t)b * NZ_ + z) * NX_ + t) * K_;
    float4 o0 = { acc[0], acc[1], acc[2], acc[3] };
    float4 o1 = { acc[4], acc[5], acc[6], acc[7] };
    reinterpret_cast<float4*>(o)[0] = o0;
    reinterpret_cast<float4*>(o)[1] = o1;
}

// ---------------- transpose rfs [B,K,NC,NS] -> [B,NC,NS,K] ----------------
// thread per (b,c,s): 8 coalesced strided reads, one 32B contiguous write.
__global__ __launch_bounds__(256) void transpose_rfs(const float* __restrict__ rfs,
                                                     float* __restrict__ rfsT) {
    size_t t = (size_t)blockIdx.x * blockDim.x + threadIdx.x;   // B*NC*NS threads
    int s = (int)(t % NS_);
    int c = (int)((t / NS_) % NC_);
    int b = (int)(t / ((size_t)NS_ * NC_));
    float v[K_];
#pragma unroll
    for (int k = 0; k < K_; ++k)
        v[k] = rfs[(((size_t)b * K_ + k) * NC_ + c) * NS_ + s];
    float* o = rfsT + (((size_t)b * NC_ + c) * NS_ + s) * K_;
    float4 o0 = { v[0], v[1], v[2], v[3] };
    float4 o1 = { v[4], v[5], v[6], v[7] };
    reinterpret_cast<float4*>(o)[0] = o0;
    reinterpret_cast<float4*>(o)[1] = o1;
}

extern "C" void kernel_launch(void* const* d_in, const int* in_sizes, int n_in,
                              void* d_out, int out_size, void* d_ws, size_t ws_size,
                              hipStream_t stream) {
    const float* rfs  = (const float*)d_in[0];
    const int*   ids  = (const int*)d_in[1];
    const float* sidx = (const float*)d_in[2];
    float*       out  = (float*)d_out;

    const size_t need = (size_t)B_ * NC_ * NS_ * K_ * sizeof(float);   // 8.4 MB
    if (ws_size >= need) {
        float* rfsT = (float*)d_ws;
        transpose_rfs<<<(B_ * NC_ * NS_) / 256, 256, 0, stream>>>(rfs, rfsT);
        das_kernel_trans<<<B_ * NZ_, NX_, 0, stream>>>(rfsT, ids, sidx, out);
    } else {
        das_kernel_plain<<<B_ * NZ_, NX_, 0, stream>>>(rfs, ids, sidx, out);
    }
}